// CIDERCriterion_45767171506361
// MI455X (gfx1250) — compile-verified
//
#include <hip/hip_runtime.h>
#include <hip/hip_bf16.h>
#include <math.h>

// Problem constants (match reference)
#define B_N 8192
#define C_N 1000
#define D_N 512
#define INV_TEMP 10.0f   // 1/TEMP ; TEMP/BASE_TEMP == 1.0

typedef float v2f __attribute__((ext_vector_type(2)));
typedef float v8f __attribute__((ext_vector_type(8)));

// ---------- wave32 reductions ----------
__device__ __forceinline__ float waveReduceAdd32(float v) {
  #pragma unroll
  for (int m = 16; m >= 1; m >>= 1) v += __shfl_xor(v, m, 32);
  return v;
}
// reductions within each 16-lane half (lanes 0-15 and 16-31 independently)
__device__ __forceinline__ float halfReduceAdd(float v) {
  #pragma unroll
  for (int m = 8; m >= 1; m >>= 1) v += __shfl_xor(v, m, 32);
  return v;
}
__device__ __forceinline__ float halfReduceMax(float v) {
  #pragma unroll
  for (int m = 8; m >= 1; m >>= 1) v = fmaxf(v, __shfl_xor(v, m, 32));
  return v;
}

__device__ __forceinline__ int imin(int a, int b) { return a < b ? a : b; }

// ---------- Phase 1: per-class sequential EMA prototype update ----------
// One wave per class; prototype row (512 f32) lives in registers (16/lane).
// Updates to the same class are applied in batch order (ballot + ffs).
__global__ void proto_update_kernel(const float* __restrict__ feat,
                                    const int* __restrict__ labels,
                                    float* __restrict__ proto) {
  const int wave = (int)((blockIdx.x * blockDim.x + threadIdx.x) >> 5);
  const int lane = (int)(threadIdx.x & 31);
  if (wave >= C_N) return;
  const int c = wave;

  float v[16];
  #pragma unroll
  for (int k = 0; k < 16; ++k) v[k] = proto[(size_t)c * D_N + k * 32 + lane];

  for (int i0 = 0; i0 < B_N; i0 += 32) {
    const int lab = labels[i0 + lane];
    unsigned long long m = __ballot(lab == c);
    while (m) {
      const int t = __ffsll(m) - 1;
      m &= (m - 1ull);
      const float* f = feat + (size_t)(i0 + t) * D_N;
      float ss = 0.f;
      #pragma unroll
      for (int k = 0; k < 16; ++k) {
        v[k] = 0.5f * v[k] + 0.5f * f[k * 32 + lane];
        ss += v[k] * v[k];
      }
      ss = waveReduceAdd32(ss);
      const float inv = 1.0f / fmaxf(sqrtf(ss), 1e-12f);
      #pragma unroll
      for (int k = 0; k < 16; ++k) v[k] *= inv;
    }
  }
  #pragma unroll
  for (int k = 0; k < 16; ++k) proto[(size_t)c * D_N + k * 32 + lane] = v[k];
}

// ---------- Phase 2a: re-normalize prototypes into pnorm (CompLoss input) ----------
__global__ void renorm_kernel(const float* __restrict__ proto,
                              float* __restrict__ pnorm) {
  const int wave = (int)((blockIdx.x * blockDim.x + threadIdx.x) >> 5);
  const int lane = (int)(threadIdx.x & 31);
  if (wave >= C_N) return;
  float v[16];
  float ss = 0.f;
  #pragma unroll
  for (int k = 0; k < 16; ++k) {
    v[k] = proto[(size_t)wave * D_N + k * 32 + lane];
    ss += v[k] * v[k];
  }
  ss = waveReduceAdd32(ss);
  const float inv = 1.0f / fmaxf(sqrtf(ss), 1e-12f);
  #pragma unroll
  for (int k = 0; k < 16; ++k)
    pnorm[(size_t)wave * D_N + k * 32 + lane] = v[k] * inv;
}

// ---------- shared GEMM micro-kernel ----------
// Computes 4 16x16 f32 accumulator tiles (acc[j]) = Arow x Brow[j]^T over K=512
// using V_WMMA_F32_16X16X4_F32.  Branch-free inner loop: pointers are
// pre-clamped, invalid lanes compute discarded values.
// Fragment layout (ISA 7.12.2): lanes 0-15 hold K = kk,kk+1; lanes 16-31 hold
// K = kk+2,kk+3  ->  per-lane base offset koff = 2*(lane>>4).
__device__ __forceinline__ void gemm_tile4(const float* __restrict__ Arow,
                                           const float* const Brow[4],
                                           int koff, v8f acc[4]) {
  for (int kk = 0; kk < D_N; kk += 16) {
    v2f a[4], b[4][4];
    #pragma unroll
    for (int u = 0; u < 4; ++u) {
      a[u] = *(const v2f*)(Arow + kk + 4 * u + koff);
      #pragma unroll
      for (int j = 0; j < 4; ++j)
        b[j][u] = *(const v2f*)(Brow[j] + kk + 4 * u + koff);
    }
    #pragma unroll
    for (int u = 0; u < 4; ++u) {
      #pragma unroll
      for (int j = 0; j < 4; ++j)
        acc[j] = __builtin_amdgcn_wmma_f32_16x16x4_f32(
            false, a[u], false, b[j][u], (short)0, acc[j], false, false);
    }
  }
}

// ---------- Phase 2b: DisLoss via fp32 WMMA ----------
// One wave per 16-row tile of proto; 4 column chunks per pass (64 cols).
__global__ void dis_loss_kernel(const float* __restrict__ P,
                                float* __restrict__ accDis) {
  const int wave = (int)((blockIdx.x * blockDim.x + threadIdx.x) >> 5);
  const int lane = (int)(threadIdx.x & 31);
  const int nTiles = (C_N + 15) / 16;  // 63
  if (wave >= nTiles) return;
  const int m0   = wave * 16;
  const int grp  = lane >> 4;      // 0: rows r, 1: rows r+8 (C/D layout)
  const int l16  = lane & 15;
  const int koff = 2 * grp;

  float rowsum[8];
  #pragma unroll
  for (int r = 0; r < 8; ++r) rowsum[r] = 0.f;

  const int arow = imin(m0 + l16, C_N - 1);        // clamped, masked later
  const float* Arow = P + (size_t)arow * D_N;

  for (int n0 = 0; n0 < 1024; n0 += 64) {          // covers cols 0..1023
    const float* Brow[4];
    #pragma unroll
    for (int j = 0; j < 4; ++j) {
      const int bcol = imin(n0 + 16 * j + l16, C_N - 1);
      Brow[j] = P + (size_t)bcol * D_N;
    }
    v8f acc[4] = {{}, {}, {}, {}};
    gemm_tile4(Arow, Brow, koff, acc);

    #pragma unroll
    for (int j = 0; j < 4; ++j) {
      const int N = n0 + 16 * j + l16;
      #pragma unroll
      for (int r = 0; r < 8; ++r) {
        const int M = m0 + r + 8 * grp;
        float e = 0.f;
        if (M < C_N && N < C_N && M != N) e = __expf(INV_TEMP * acc[j][r]);
        rowsum[r] += e;
      }
    }
  }
  float contrib = 0.f;
  #pragma unroll
  for (int r = 0; r < 8; ++r) {
    const float s = halfReduceAdd(rowsum[r]);
    const int M = m0 + r + 8 * grp;
    if (l16 == 0 && M < C_N) contrib += logf(s / (float)(C_N - 1));
  }
  if (l16 == 0) atomicAdd(accDis, contrib);
}

// ---------- Phase 3: CompLoss via fp32 WMMA, fused online softmax ----------
// One wave per 16-row tile of features (512 tiles total); 4 col chunks/pass.
__global__ void comp_loss_kernel(const float* __restrict__ F,
                                 const float* __restrict__ P,
                                 const int* __restrict__ labels,
                                 float* __restrict__ accComp) {
  const int wave = (int)((blockIdx.x * blockDim.x + threadIdx.x) >> 5);
  const int lane = (int)(threadIdx.x & 31);
  if (wave >= B_N / 16) return;
  const int m0   = wave * 16;
  const int grp  = lane >> 4;
  const int l16  = lane & 15;
  const int koff = 2 * grp;

  float mrun[8], srun[8], lv[8];
  int lab[8];
  #pragma unroll
  for (int r = 0; r < 8; ++r) {
    mrun[r] = -__builtin_inff();
    srun[r] = 0.f;
    lv[r]   = 0.f;
    lab[r]  = labels[m0 + r + 8 * grp];
  }

  const float* Arow = F + (size_t)(m0 + l16) * D_N;  // all feature rows valid

  for (int n0 = 0; n0 < 1024; n0 += 64) {            // covers cols 0..1023
    const float* Brow[4];
    #pragma unroll
    for (int j = 0; j < 4; ++j) {
      const int bcol = imin(n0 + 16 * j + l16, C_N - 1);
      Brow[j] = P + (size_t)bcol * D_N;
    }
    v8f acc[4] = {{}, {}, {}, {}};
    gemm_tile4(Arow, Brow, koff, acc);

    #pragma unroll
    for (int j = 0; j < 4; ++j) {
      const int N = n0 + 16 * j + l16;
      #pragma unroll
      for (int r = 0; r < 8; ++r) {
        const float l  = (N < C_N) ? INV_TEMP * acc[j][r] : -1e30f;
        const float cm = halfReduceMax(l);
        const float nm = fmaxf(mrun[r], cm);
        const float e  = (N < C_N) ? __expf(l - nm) : 0.f;
        const float cs = halfReduceAdd(e);
        srun[r] = srun[r] * __expf(mrun[r] - nm) + cs;
        mrun[r] = nm;
        if (N == lab[r]) lv[r] += l;   // raw logit at label column
      }
    }
  }
  float sumval = 0.f;
  #pragma unroll
  for (int r = 0; r < 8; ++r) {
    const float lvr = halfReduceAdd(lv[r]);
    const float val = lvr - (mrun[r] + logf(srun[r]));  // log_prob at label
    if (l16 == 0) sumval += val;
  }
  if (l16 == 0) atomicAdd(accComp, sumval);
}

// ---------- Finalize ----------
__global__ void finalize_kernel(const float* __restrict__ accs,
                                float* __restrict__ out) {
  const float comp = -accs[0] / (float)B_N;   // -(TEMP/BASE_TEMP)*mean
  const float dis  =  accs[1] / (float)C_N;   //  (TEMP/BASE_TEMP)*mean
  out[0] = 1.0f * comp + dis;                 // W * comp + dis
}

extern "C" void kernel_launch(void* const* d_in, const int* in_sizes, int n_in,
                              void* d_out, int out_size, void* d_ws, size_t ws_size,
                              hipStream_t stream) {
  const float* features   = (const float*)d_in[0];
  const float* prototypes = (const float*)d_in[1];
  const int*   labels     = (const int*)d_in[2];
  float* out = (float*)d_out;

  float* wsf       = (float*)d_ws;
  float* acc       = wsf;                       // [0]=comp sum, [1]=dis sum
  float* workproto = wsf + 16;                  // C*D floats
  float* pnorm     = workproto + (size_t)C_N * D_N;

  // zero accumulators; copy prototypes into mutable workspace (capture-safe)
  hipMemsetAsync(acc, 0, 16 * sizeof(float), stream);
  hipMemcpyAsync(workproto, prototypes, (size_t)C_N * D_N * sizeof(float),
                 hipMemcpyDeviceToDevice, stream);

  // 1000 waves (one per class), 8 waves/block
  proto_update_kernel<<<125, 256, 0, stream>>>(features, labels, workproto);
  // 1000 waves (one per prototype row)
  renorm_kernel<<<125, 256, 0, stream>>>(workproto, pnorm);
  // 63 row-tiles of proto -> 64 waves
  dis_loss_kernel<<<8, 256, 0, stream>>>(workproto, acc + 1);
  // 512 row-tiles of features -> 512 waves
  comp_loss_kernel<<<64, 256, 0, stream>>>(features, pnorm, labels, acc + 0);

  finalize_kernel<<<1, 1, 0, stream>>>(acc, out);
}